// PointLossFunction_17145509446220
// MI455X (gfx1250) — compile-verified
//
#include <hip/hip_runtime.h>
#include <math.h>

// ---------------------------------------------------------------------------
// Sinkhorn divergence loss (geomloss-style, debiased) for 80x80x5 point clouds
// on MI455X / gfx1250.  Core op: fused [6400x8]x[8x6400] fp32 WMMA GEMM +
// streaming row-wise logsumexp (softmin), repeated over the eps schedule.
// Column panels are streamed into LDS by the Tensor Data Mover (TDM), using
// its pad feature (1 DWORD after every 8) to produce the stride-9 bank-
// conflict-free layout, double-buffered against compute via TENSORcnt.
// Final answer uses the envelope-theorem gradient:
//     grad_i = bary(x,x,p)_i - bary(x,y,g)_i          (RES^2/N == 1)
//     loss   = mean(grad^2) over 6400*5 elements.
// ---------------------------------------------------------------------------

typedef __attribute__((ext_vector_type(2))) float v2f;
typedef __attribute__((ext_vector_type(8))) float v8f;
typedef __attribute__((ext_vector_type(4))) unsigned int v4u;
typedef __attribute__((ext_vector_type(8))) int v8i;
typedef __attribute__((ext_vector_type(4))) int v4i;

#define NPTS   6400
#define IMGH   80
#define PANEL  256          // columns staged per LDS panel
#define WAVES  8            // wave32s per block
#define ROWS_PER_WAVE 16
#define ROWS_PER_BLOCK (WAVES * ROWS_PER_WAVE)      // 128
#define NBLK_SOFT (NPTS / ROWS_PER_BLOCK)           // 50
#define LOGW   (-8.7640533f)                        // log(1/6400)

static __device__ __forceinline__ float clamp01(float x) {
    return fminf(fmaxf(x, 0.0f), 1.0f);
}

// ---------------------------------------------------------------------------
// TDM: DMA one 256-column panel (2048 f32, contiguous) from global into LDS,
// inserting 1 pad DWORD after every 8 -> 9-float LDS stride per column.
// D# layout per CDNA5 ISA ch.8 (group0 128b, group1 256b; groups 2/3 unused).
// This toolchain exposes the 6-arg builtin: (g0, g1, g2, g3, g_extra, cpol).
// ---------------------------------------------------------------------------
static __device__ __forceinline__ void tdm_issue_panel(const float* gsrc,
                                                       unsigned lds_addr) {
    unsigned long long ga = (unsigned long long)(uintptr_t)gsrc;
    v4u g0;
    g0[0] = 1u;                                   // count=1, user descriptor
    g0[1] = lds_addr;                             // lds_addr (bytes)
    g0[2] = (unsigned)(ga & 0xffffffffu);         // global_addr[31:0]
    g0[3] = (unsigned)((ga >> 32) & 0x01ffffffu)  // global_addr[56:32]
          | (2u << 30);                           // type=2 ("image")
    v8i g1;
    g1[0] = (2 << 16)                 // data_size: 4 bytes
          | (1 << 20)                 // pad_enable
          | (2 << 22)                 // pad_interval: every 8 DWORDs
          | (0 << 25);                // pad_amount: 1 DWORD
    g1[1] = (int)((PANEL * 8) << 16); // tensor_dim0[15:0] = 2048
    g1[2] = (1 << 16);                // tensor_dim0 hi = 0 | tensor_dim1 = 1
    g1[3] = (int)((PANEL * 8) << 16); // tensor_dim1 hi = 0 | tile_dim0 = 2048
    g1[4] = 1;                        // tile_dim1 = 1, tile_dim2 = 0
    g1[5] = PANEL * 8;                // tensor_dim0_stride = 2048
    g1[6] = 0;
    g1[7] = 0;
    v4i z4 = {0, 0, 0, 0};
    v8i z8 = {0, 0, 0, 0, 0, 0, 0, 0};
    __builtin_amdgcn_tensor_load_to_lds(g0, g1, z4, z4, z8, 0);
}

// ---------------------------------------------------------------------------
// Build padded point arrays X (render5d, clipped) and Y (target) plus half
// squared norms.  Layout: [i*8 + k], k=0..4 features, k=5..7 zero pad.
// ---------------------------------------------------------------------------
__global__ __launch_bounds__(256) void build_points_kernel(
    const float* __restrict__ pos, const float* __restrict__ images,
    const float* __restrict__ gt_rgb, const unsigned char* __restrict__ haspos,
    float* __restrict__ X, float* __restrict__ Y,
    float* __restrict__ sx, float* __restrict__ sy)
{
    int i = blockIdx.x * blockDim.x + threadIdx.x;
    if (i >= NPTS) return;
    int r = i / IMGH, c = i % IMGH;
    float u = (float)c * (1.0f / 80.0f);    // uv[...,0] = w/RES
    float v = (float)r * (1.0f / 80.0f);    // uv[...,1] = h/RES
    bool hp = haspos[i] != 0;
    float ru0 = hp ? (pos[2 * i + 0] + 1.0f) * 0.5f : u;
    float ru1 = hp ? (pos[2 * i + 1] + 1.0f) * 0.5f : v;

    float xv[5], yv[5];
    xv[0] = clamp01(images[3 * i + 0]);
    xv[1] = clamp01(images[3 * i + 1]);
    xv[2] = clamp01(images[3 * i + 2]);
    xv[3] = clamp01(ru0);                   // xm clips all 5 dims
    xv[4] = clamp01(ru1);
    yv[0] = clamp01(gt_rgb[3 * i + 0]);
    yv[1] = clamp01(gt_rgb[3 * i + 1]);
    yv[2] = clamp01(gt_rgb[3 * i + 2]);
    yv[3] = ru0;                            // target uv is NOT clipped
    yv[4] = ru1;

    float nx = 0.0f, ny = 0.0f;
#pragma unroll
    for (int k = 0; k < 5; ++k) {
        X[i * 8 + k] = xv[k];
        Y[i * 8 + k] = yv[k];
        nx = fmaf(xv[k], xv[k], nx);
        ny = fmaf(yv[k], yv[k], ny);
    }
#pragma unroll
    for (int k = 5; k < 8; ++k) { X[i * 8 + k] = 0.0f; Y[i * 8 + k] = 0.0f; }
    sx[i] = 0.5f * nx;
    sy[i] = 0.5f * ny;
}

__global__ __launch_bounds__(256) void init_potentials_kernel(
    float* __restrict__ f, float* __restrict__ g, float* __restrict__ p)
{
    int i = blockIdx.x * blockDim.x + threadIdx.x;
    if (i < NPTS) { f[i] = 0.0f; g[i] = 0.0f; p[i] = 0.0f; }
}

// ---------------------------------------------------------------------------
// softmin pass:
//   out[i] = -eps * logsumexp_j( logw + (pot_j - 0.5||y_j||^2)/eps
//                                - 0.5||x_i||^2/eps + (x_i . y_j)/eps )
// Each wave owns 16 rows; columns streamed through LDS in 256-wide panels,
// DMA'd by the TDM (double-buffered, TENSORcnt-tracked).  Dot products via
// V_WMMA_F32_16X16X4_F32 (two per tile: K=0..3, K=4..7).  BARY variant also
// accumulates the transport barycenter sum_j softmax_ij * y_j.
// ---------------------------------------------------------------------------
template <bool BARY>
__global__ __launch_bounds__(256) void softmin_kernel(
    const float* __restrict__ Xr,  const float* __restrict__ snR,
    const float* __restrict__ Yc,  const float* __restrict__ snC,
    const float* __restrict__ pot, float eps,
    float* __restrict__ outFt, float* __restrict__ outBary)
{
    __shared__ float sYs[2][PANEL * 9];   // stride-9: conflict-free B reads
    __shared__ float sHH[2][PANEL];

    const float inv_eps = 1.0f / eps;
    const int tid   = threadIdx.x;
    const int lane  = tid & 31;
    const int wave  = tid >> 5;
    const int laneM = lane & 15;
    const int kb    = (lane >> 4) << 1;   // A/B VGPR half: K base 0 or 2
    const int rowAdd = (lane >> 4) << 3;  // C/D row offset: 0 or 8
    const int row0 = (blockIdx.x * WAVES + wave) * ROWS_PER_WAVE;

    // A operands (16x4 fp32 per WMMA), loaded once per wave.
    const float* ax = &Xr[(row0 + laneM) * 8];
    v2f a0; a0.x = ax[kb];     a0.y = ax[kb + 1];
    v2f a1; a1.x = ax[4 + kb]; a1.y = ax[5 + kb];

    // Row half-norms / eps, in the C/D-matrix row mapping.
    float snr[8];
#pragma unroll
    for (int r = 0; r < 8; ++r)
        snr[r] = snR[row0 + r + rowAdd] * inv_eps;

    float m[8], s[8];
    float w[BARY ? 40 : 1];
#pragma unroll
    for (int r = 0; r < 8; ++r) { m[r] = -__builtin_inff(); s[r] = 0.0f; }
    if (BARY) {
#pragma unroll
        for (int j = 0; j < (BARY ? 40 : 1); ++j) w[j] = 0.0f;
    }

    const int NPAN = NPTS / PANEL;        // 25 panels

    // Prologue: DMA panel 0 into buffer 0, stage its hh in parallel.
    if (wave == 0)
        tdm_issue_panel(&Yc[0], (unsigned)(uintptr_t)&sYs[0][0]);
    sHH[0][tid] = LOGW + (pot[tid] - snC[tid]) * inv_eps;

    for (int pp = 0; pp < NPAN; ++pp) {
        const int buf = pp & 1;
        if (pp + 1 < NPAN) {
            if (wave == 0) {
                // Issue next panel into the other buffer, then wait for the
                // current one (in-order per wave: cnt<=1 => oldest done).
                tdm_issue_panel(&Yc[(pp + 1) * (PANEL * 8)],
                                (unsigned)(uintptr_t)&sYs[buf ^ 1][0]);
                __builtin_amdgcn_s_wait_tensorcnt(1);
            }
            int c = (pp + 1) * PANEL + tid;
            sHH[buf ^ 1][tid] = LOGW + (pot[c] - snC[c]) * inv_eps;
            if (pp + 2 < NPAN)
                __builtin_prefetch(&pot[c + PANEL], 0, 1); // global_prefetch_b8
        } else {
            if (wave == 0) __builtin_amdgcn_s_wait_tensorcnt(0);
        }
        __syncthreads();   // panel `buf` visible to all waves

#pragma unroll 2
        for (int t = 0; t < PANEL / 16; ++t) {
            const float* ys = &sYs[buf][(t * 16 + laneM) * 9];
            v2f b0; b0.x = ys[kb];     b0.y = ys[kb + 1];
            v2f b1; b1.x = ys[4 + kb]; b1.y = ys[5 + kb];

            v8f acc = {};
            acc = __builtin_amdgcn_wmma_f32_16x16x4_f32(
                false, a0, false, b0, (short)0, acc, false, false);
            acc = __builtin_amdgcn_wmma_f32_16x16x4_f32(
                false, a1, false, b1, (short)0, acc, false, false);

            float hh = sHH[buf][t * 16 + laneM];
            float yv[5];
            if (BARY) {
#pragma unroll
                for (int k = 0; k < 5; ++k) yv[k] = ys[k];
            }
#pragma unroll
            for (int r = 0; r < 8; ++r) {
                float z  = fmaf(acc[r], inv_eps, hh - snr[r]);
                float mo = m[r];
                float mn = fmaxf(mo, z);
                float eo = __expf(mo - mn);
                float en = __expf(z - mn);
                s[r] = fmaf(s[r], eo, en);
                m[r] = mn;
                if (BARY) {
#pragma unroll
                    for (int k = 0; k < 5; ++k)
                        w[r * 5 + k] = fmaf(w[r * 5 + k], eo, en * yv[k]);
                }
            }
        }
        __syncthreads();   // all waves done with `buf` before it is re-DMA'd
    }

    // Combine the 16 column-slots of each row across lanes (halves disjoint).
#pragma unroll
    for (int off = 1; off < 16; off <<= 1) {
#pragma unroll
        for (int r = 0; r < 8; ++r) {
            float mo = __shfl_xor(m[r], off, 32);
            float so = __shfl_xor(s[r], off, 32);
            float mn = fmaxf(m[r], mo);
            float fs = __expf(m[r] - mn);
            float fo = __expf(mo - mn);
            s[r] = s[r] * fs + so * fo;
            if (BARY) {
#pragma unroll
                for (int k = 0; k < 5; ++k) {
                    float wo = __shfl_xor(w[r * 5 + k], off, 32);
                    w[r * 5 + k] = w[r * 5 + k] * fs + wo * fo;
                }
            }
            m[r] = mn;
        }
    }

    if (laneM == 0) {
#pragma unroll
        for (int r = 0; r < 8; ++r) {
            int row = row0 + r + rowAdd;
            if (!BARY) {
                outFt[row] = -eps * (m[r] + __logf(s[r]));
            } else {
                float invs = 1.0f / s[r];
#pragma unroll
                for (int k = 0; k < 5; ++k)
                    outBary[row * 8 + k] = w[r * 5 + k] * invs;
            }
        }
    }
}

// Symmetric eps-annealing update (uses OLD f,g,p on the RHS by construction).
__global__ __launch_bounds__(256) void update_kernel(
    float* __restrict__ f, const float* __restrict__ ft,
    float* __restrict__ g, const float* __restrict__ gt,
    float* __restrict__ p, const float* __restrict__ pt)
{
    int i = blockIdx.x * blockDim.x + threadIdx.x;
    if (i < NPTS) {
        f[i] = 0.5f * (f[i] + ft[i]);
        g[i] = 0.5f * (g[i] + gt[i]);
        p[i] = 0.5f * (p[i] + pt[i]);
    }
}

// loss = mean over 6400*5 of (bx - by)^2   (RES^2/N == 1 so no extra scale)
__global__ __launch_bounds__(256) void loss_kernel(
    const float* __restrict__ bx, const float* __restrict__ by,
    float* __restrict__ out)
{
    __shared__ float red[256];
    float acc = 0.0f;
    for (int i = threadIdx.x; i < NPTS; i += 256) {
#pragma unroll
        for (int k = 0; k < 5; ++k) {
            float d = bx[i * 8 + k] - by[i * 8 + k];
            acc = fmaf(d, d, acc);
        }
    }
    red[threadIdx.x] = acc;
    __syncthreads();
    for (int stride = 128; stride > 0; stride >>= 1) {
        if (threadIdx.x < stride) red[threadIdx.x] += red[threadIdx.x + stride];
        __syncthreads();
    }
    if (threadIdx.x == 0) out[0] = red[0] * (1.0f / (NPTS * 5.0f));
}

// ---------------------------------------------------------------------------
extern "C" void kernel_launch(void* const* d_in, const int* in_sizes, int n_in,
                              void* d_out, int out_size, void* d_ws, size_t ws_size,
                              hipStream_t stream) {
    (void)in_sizes; (void)n_in; (void)out_size; (void)ws_size;
    const float* pos            = (const float*)d_in[0];
    const float* images         = (const float*)d_in[1];
    const float* gt_rgb         = (const float*)d_in[2];
    const unsigned char* haspos = (const unsigned char*)d_in[3];

    float* ws = (float*)d_ws;
    float* X   = ws;                    // 6400*8
    float* Y   = ws + 51200;            // 6400*8
    float* sx  = ws + 102400;           // 6400
    float* sy  = ws + 108800;
    float* f   = ws + 115200;
    float* g   = ws + 121600;
    float* p   = ws + 128000;
    float* ft  = ws + 134400;
    float* gtv = ws + 140800;
    float* pt  = ws + 147200;
    float* by  = ws + 153600;           // 6400*8 (bary of OT(x,y))
    float* bx  = ws + 204800;           // 6400*8 (bary of OT(x,x))

    build_points_kernel<<<(NPTS + 255) / 256, 256, 0, stream>>>(
        pos, images, gt_rgb, haspos, X, Y, sx, sy);
    init_potentials_kernel<<<(NPTS + 255) / 256, 256, 0, stream>>>(f, g, p);

    // geomloss eps schedule: diam^2 -> blur^2 by scaling^2, then 3x blur^2
    float eps_list[16]; int ne = 0;
    double e = 5.0;
    while (e > 1e-4) { eps_list[ne++] = (float)e; e *= 0.25; }
    for (int i = 0; i < 3; ++i) eps_list[ne++] = 1e-4f;

    for (int it = 0; it < ne; ++it) {
        float eps = eps_list[it];
        softmin_kernel<false><<<NBLK_SOFT, 256, 0, stream>>>(
            X, sx, Y, sy, g, eps, ft, nullptr);          // ft uses old g
        softmin_kernel<false><<<NBLK_SOFT, 256, 0, stream>>>(
            Y, sy, X, sx, f, eps, gtv, nullptr);         // gt uses old f
        softmin_kernel<false><<<NBLK_SOFT, 256, 0, stream>>>(
            X, sx, X, sx, p, eps, pt, nullptr);          // self potential
        update_kernel<<<(NPTS + 255) / 256, 256, 0, stream>>>(
            f, ft, g, gtv, p, pt);
        // q potential is skipped: it only feeds Gyy, whose gradient wrt xm is 0
    }

    const float eps_f = 1e-4f;
    softmin_kernel<true><<<NBLK_SOFT, 256, 0, stream>>>(
        X, sx, Y, sy, g, eps_f, nullptr, by);            // transport bary of F
    softmin_kernel<true><<<NBLK_SOFT, 256, 0, stream>>>(
        X, sx, X, sx, p, eps_f, nullptr, bx);            // transport bary of Fxx
    loss_kernel<<<1, 256, 0, stream>>>(bx, by, (float*)d_out);
}